// GAT_70506183131634
// MI455X (gfx1250) — compile-verified
//
#include <hip/hip_runtime.h>
#include <math.h>

// ---------------------------------------------------------------------------
// GAT layer, algebraically refactored:
//   v = W_proj^T @ W_key[0,   0:128]     (128 floats)
//   u = W_proj^T @ W_key[0, 128:256]     (128 floats)
//   a_i = h_i . v                         (per node)
//   b_j = hjs_j . u                       (per edge)
//   e_j = leaky_relu(a_seg(j) + b_j); softmax per node (online, 1 pass)
//   g_i = sum_j att_j * hjs_j             (staged in d_out)
//   out = relu(G @ W_proj^T)              (WMMA f32 16x16x4, in-place)
// Memory-bound: ~512 MB total traffic -> ~22 us at 23.3 TB/s.
// ---------------------------------------------------------------------------

typedef __attribute__((ext_vector_type(2))) float v2f;
typedef __attribute__((ext_vector_type(4))) float f4;   // for nontemporal loads
typedef __attribute__((ext_vector_type(8))) float v8f;

#define Dk 128

__global__ void gat_precompute_kernel(const float* __restrict__ W_proj,
                                      const float* __restrict__ W_key,
                                      float* __restrict__ vu) {
    int k = threadIdx.x;                       // 128 threads
    float sv = 0.f, su = 0.f;
    for (int d = 0; d < Dk; ++d) {
        float w = W_proj[d * Dk + k];          // W_proj[d][k]
        sv = fmaf(w, W_key[d], sv);
        su = fmaf(w, W_key[Dk + d], su);
    }
    vu[k]      = sv;
    vu[Dk + k] = su;
}

// Single-block 3-phase scan: serial per-thread sums, one 10-step block scan,
// serial per-thread offset writes. ~20 barriers total (vs ~1000 before).
__global__ void gat_scan_kernel(const int* __restrict__ n_list,
                                int* __restrict__ offsets, int N) {
    __shared__ int sums[1024];
    const int tid   = threadIdx.x;
    const int chunk = (N + 1023) >> 10;
    int lo = tid * chunk;       if (lo > N) lo = N;
    int hi = lo + chunk;        if (hi > N) hi = N;

    int s = 0;
    for (int i = lo; i < hi; ++i) s += n_list[i];
    sums[tid] = s;
    __syncthreads();
    for (int o = 1; o < 1024; o <<= 1) {       // Hillis-Steele inclusive scan
        int t = (tid >= o) ? sums[tid - o] : 0;
        __syncthreads();
        sums[tid] += t;
        __syncthreads();
    }
    int run = sums[tid] - s;                   // exclusive prefix of this slice
    for (int i = lo; i < hi; ++i) { offsets[i] = run; run += n_list[i]; }
    if (tid == 1023) offsets[N] = sums[1023];
}

// One wave32 per node: wave-wide dots + online softmax, single NT pass over hjs.
// Edges processed 4 at a time for memory-level parallelism (4 outstanding
// 512B row loads, pipelined shuffle reductions).
__global__ void __launch_bounds__(256)
gat_node_attn_kernel(const float* __restrict__ h,
                     const float* __restrict__ hjs,
                     const int* __restrict__ n_list,
                     const int* __restrict__ offsets,
                     const float* __restrict__ vu,
                     float* __restrict__ G, int N) {
    const int lane = threadIdx.x & 31;                       // wave32 (gfx1250)
    const int node = (blockIdx.x * blockDim.x + threadIdx.x) >> 5;
    if (node >= N) return;

    // each lane owns 4 contiguous floats of the 128-wide rows
    const f4 v4 = ((const f4*)vu)[lane];
    const f4 u4 = ((const f4*)(vu + Dk))[lane];
    const f4 h4 = __builtin_nontemporal_load(&((const f4*)(h + (size_t)node * Dk))[lane]);

    float ap = fmaf(h4.x, v4.x, fmaf(h4.y, v4.y, fmaf(h4.z, v4.z, h4.w * v4.w)));
    #pragma unroll
    for (int o = 16; o >= 1; o >>= 1) ap += __shfl_xor(ap, o, 32);
    const float aN = ap;                                     // h_i . v, all lanes

    const int off0 = offsets[node];
    const int deg  = n_list[node];

    float m = -INFINITY, s = 0.f;
    float gx = 0.f, gy = 0.f, gz = 0.f, gw = 0.f;

    auto upd = [&](float bp, const f4& r4) {
        float e = aN + bp;
        e = (e > 0.f) ? e : 0.01f * e;                       // leaky_relu, slope .01
        float mn    = fmaxf(m, e);
        float scale = __expf(m - mn);                        // exp(-inf)=0 first time
        float w     = __expf(e - mn);
        s  = fmaf(s, scale, w);
        gx = fmaf(gx, scale, w * r4.x);
        gy = fmaf(gy, scale, w * r4.y);
        gz = fmaf(gz, scale, w * r4.z);
        gw = fmaf(gw, scale, w * r4.w);
        m = mn;
    };
    auto row = [&](int j) -> f4 {
        return __builtin_nontemporal_load(
            &((const f4*)(hjs + (size_t)(off0 + j) * Dk))[lane]);   // TH=NT stream
    };

    int j = 0;
    for (; j + 4 <= deg; j += 4) {
        if (j + 8 <= deg)                                    // global_prefetch_b8
            __builtin_prefetch(hjs + (size_t)(off0 + j + 4) * Dk + lane * 4, 0, 0);
        const f4 r0 = row(j), r1 = row(j + 1), r2 = row(j + 2), r3 = row(j + 3);
        float b0 = fmaf(r0.x, u4.x, fmaf(r0.y, u4.y, fmaf(r0.z, u4.z, r0.w * u4.w)));
        float b1 = fmaf(r1.x, u4.x, fmaf(r1.y, u4.y, fmaf(r1.z, u4.z, r1.w * u4.w)));
        float b2 = fmaf(r2.x, u4.x, fmaf(r2.y, u4.y, fmaf(r2.z, u4.z, r2.w * u4.w)));
        float b3 = fmaf(r3.x, u4.x, fmaf(r3.y, u4.y, fmaf(r3.z, u4.z, r3.w * u4.w)));
        #pragma unroll
        for (int o = 16; o >= 1; o >>= 1) {                  // 4 independent trees
            b0 += __shfl_xor(b0, o, 32);
            b1 += __shfl_xor(b1, o, 32);
            b2 += __shfl_xor(b2, o, 32);
            b3 += __shfl_xor(b3, o, 32);
        }
        upd(b0, r0); upd(b1, r1); upd(b2, r2); upd(b3, r3);
    }
    for (; j < deg; ++j) {
        const f4 r0 = row(j);
        float b0 = fmaf(r0.x, u4.x, fmaf(r0.y, u4.y, fmaf(r0.z, u4.z, r0.w * u4.w)));
        #pragma unroll
        for (int o = 16; o >= 1; o >>= 1) b0 += __shfl_xor(b0, o, 32);
        upd(b0, r0);
    }

    const float inv = 1.f / s;                               // deg >= 8, s > 0
    f4 o4; o4.x = gx * inv; o4.y = gy * inv; o4.z = gz * inv; o4.w = gw * inv;
    ((f4*)(G + (size_t)node * Dk))[lane] = o4;               // cached: reread next
}

// out = relu(G @ W_proj^T), in place on G. One block = 16 rows x 128 cols,
// 8 waves, each wave computes a 16x16 tile via V_WMMA_F32_16X16X4_F32.
__global__ void __launch_bounds__(256)
gat_final_proj_kernel(const float* __restrict__ Wp, float* __restrict__ G, int N) {
    extern __shared__ float smem[];
    float* Wlds = smem;                 // 128*128 floats (64 KB): W_proj row-major
    float* Glds = smem + Dk * Dk;       // 16*128 floats (8 KB): row tile

    const int tid  = threadIdx.x;
    const int row0 = blockIdx.x * 16;

    for (int idx = tid; idx < (Dk * Dk) / 4; idx += 256)
        ((float4*)Wlds)[idx] = ((const float4*)Wp)[idx];
    for (int idx = tid; idx < (16 * Dk) / 4; idx += 256) {
        int r = idx >> 5, c4 = idx & 31;
        int gr = row0 + r;
        float4 val = make_float4(0.f, 0.f, 0.f, 0.f);
        if (gr < N) val = ((const float4*)(G + (size_t)gr * Dk))[c4];
        ((float4*)Glds)[idx] = val;
    }
    __syncthreads();

    const int wave  = tid >> 5;
    const int lane  = tid & 31;
    const int col0  = wave * 16;        // this wave's 16 output columns
    const int lm    = lane & 15;
    const int khalf = (lane >> 4) << 1; // lanes 0-15 -> K 0,1 ; lanes 16-31 -> K 2,3

    v8f c = {};                         // fp32 accumulator tile
    #pragma unroll 4
    for (int k0 = 0; k0 < Dk; k0 += 4) {
        // A (16x4): A[m][k] = G[row0+m][k0+k]
        v2f a, b;
        a.x = Glds[lm * Dk + k0 + khalf];
        a.y = Glds[lm * Dk + k0 + khalf + 1];
        // B (4x16): B[k][n] = Wp[col0+n][k0+k]   (out = G @ Wp^T)
        b.x = Wlds[(col0 + lm) * Dk + k0 + khalf];
        b.y = Wlds[(col0 + lm) * Dk + k0 + khalf + 1];
        c = __builtin_amdgcn_wmma_f32_16x16x4_f32(
                /*neg_a=*/false, a, /*neg_b=*/false, b,
                /*c_mod=*/(short)0, c, /*reuse_a=*/false, /*reuse_b=*/false);
    }
    __syncthreads();                    // done reading Glds; reuse it for output

    #pragma unroll
    for (int r = 0; r < 8; ++r) {       // C/D layout: VGPR r -> row r (+8 hi lanes)
        int mrow = r + ((lane >= 16) ? 8 : 0);
        float val = c[r];
        Glds[mrow * Dk + col0 + lm] = (val > 0.f) ? val : 0.f;   // relu
    }
    __syncthreads();

    for (int idx = tid; idx < (16 * Dk) / 4; idx += 256) {
        int r = idx >> 5, c4 = idx & 31;
        int gr = row0 + r;
        if (gr < N) ((float4*)(G + (size_t)gr * Dk))[c4] = ((float4*)Glds)[idx];
    }
}

extern "C" void kernel_launch(void* const* d_in, const int* in_sizes, int n_in,
                              void* d_out, int out_size, void* d_ws, size_t ws_size,
                              hipStream_t stream) {
    const float* h      = (const float*)d_in[0];
    const float* hjs    = (const float*)d_in[1];
    const int*   n_list = (const int*)d_in[2];
    const float* W_proj = (const float*)d_in[3];
    const float* W_key  = (const float*)d_in[4];
    const int N = in_sizes[2];          // 50000 nodes

    float* vu      = (float*)d_ws;                                // 256 floats
    int*   offsets = (int*)((char*)d_ws + 256 * sizeof(float));   // N+1 ints
    float* G       = (float*)d_out;     // staging for g_i, then final output

    gat_precompute_kernel<<<1, 128, 0, stream>>>(W_proj, W_key, vu);
    gat_scan_kernel<<<1, 1024, 0, stream>>>(n_list, offsets, N);

    const int blocks2 = (N + 7) / 8;    // 8 wave32 per 256-thread block
    gat_node_attn_kernel<<<blocks2, 256, 0, stream>>>(h, hjs, n_list, offsets, vu, G, N);

    const int blocks3 = (N + 15) / 16;
    const size_t lds = (size_t)(Dk * Dk + 16 * Dk) * sizeof(float);  // 72 KB
    gat_final_proj_kernel<<<blocks3, 256, lds, stream>>>(W_proj, G, N);
}